// GAT_60335700574518
// MI455X (gfx1250) — compile-verified
//
#include <hip/hip_runtime.h>
#include <hip/hip_bf16.h>
#include <math.h>
#include <stdint.h>

typedef float v2f __attribute__((ext_vector_type(2)));
typedef float v4f __attribute__((ext_vector_type(4)));
typedef float v8f __attribute__((ext_vector_type(8)));

#define H_DIM     64
#define B_GRAPHS  128
#define C_CLASSES 10
#define NEG_SLOPE 0.2f

// 1 = stage W into LDS with gfx1250 async-copy (global_load_async_to_lds_b128
//     + s_wait_asynccnt); 0 = plain global->LDS staging.
#define GAT_ASYNC_STAGE 1

// ---------------------------------------------------------------------------
// GEMM: Hout[N,64] = X[N,K] @ W[64,K]^T using V_WMMA_F32_16X16X4_F32.
// W is staged into LDS once per block (<=32KB), so the K-loop reads
// B-fragments via ds_load_b64 and only streams A-fragments from HBM.
// One wave computes a 16x64 row strip (4 col tiles), K-loop step 4.
// A-frag (16x4 f32): lane L<16 -> row L, k={k0,k0+1}; lane>=16 -> k={k0+2,k0+3}
// B-frag (4x16 f32): same lane pattern, N across lanes.
// D (16x16 f32): VGPR r -> row r (lanes 0-15) / row r+8 (lanes 16-31).
// Requires N % 16 == 0 (true: 100000 = 16*6250).
// ---------------------------------------------------------------------------
template <int K>
__global__ __launch_bounds__(256) void gat_gemm_wmma_f32(
    const float* __restrict__ X, const float* __restrict__ W,
    float* __restrict__ Hout, int N) {
  __shared__ float Wlds[H_DIM * K];

  // ---- cooperative staging of W[64,K] into LDS ----
  {
    const int total4 = (H_DIM * K) / 4;         // float4 units
    for (int i = threadIdx.x; i < total4; i += blockDim.x) {
      const float* g = W + (size_t)i * 4;
#if GAT_ASYNC_STAGE
      uint32_t laddr = (uint32_t)(size_t)&Wlds[i * 4];  // LDS byte offset
      asm volatile("global_load_async_to_lds_b128 %0, %1, off"
                   :: "v"(laddr), "v"(g) : "memory");
#else
      *(v4f*)&Wlds[i * 4] = *(const v4f*)g;
#endif
    }
#if GAT_ASYNC_STAGE
    asm volatile("s_wait_asynccnt 0x0" ::: "memory");
#endif
    __syncthreads();
  }

  const int lane = threadIdx.x & 31;
  const int wave = (blockIdx.x * blockDim.x + threadIdx.x) >> 5;
  const int rowb = wave * 16;
  if (rowb >= N) return;

  const int idx15 = lane & 15;            // A: row-in-tile, B: col-in-tile
  const int khalf = (lane >> 4) << 1;     // 0 or 2

  v8f acc0 = {}; v8f acc1 = {}; v8f acc2 = {}; v8f acc3 = {};
  const float* xr = X + (size_t)(rowb + idx15) * K + khalf;
  const float* l0 = &Wlds[( 0 + idx15) * K + khalf];
  const float* l1 = &Wlds[(16 + idx15) * K + khalf];
  const float* l2 = &Wlds[(32 + idx15) * K + khalf];
  const float* l3 = &Wlds[(48 + idx15) * K + khalf];

#pragma unroll
  for (int k0 = 0; k0 < K; k0 += 4) {
    v2f a  = *(const v2f*)(xr + k0);
    v2f b0 = *(const v2f*)(l0 + k0);
    v2f b1 = *(const v2f*)(l1 + k0);
    v2f b2 = *(const v2f*)(l2 + k0);
    v2f b3 = *(const v2f*)(l3 + k0);
    acc0 = __builtin_amdgcn_wmma_f32_16x16x4_f32(false, a, false, b0, (short)0, acc0, false, false);
    acc1 = __builtin_amdgcn_wmma_f32_16x16x4_f32(false, a, false, b1, (short)0, acc1, false, false);
    acc2 = __builtin_amdgcn_wmma_f32_16x16x4_f32(false, a, false, b2, (short)0, acc2, false, false);
    acc3 = __builtin_amdgcn_wmma_f32_16x16x4_f32(false, a, false, b3, (short)0, acc3, false, false);
  }

  const int rbase = rowb + ((lane >> 4) << 3);
#pragma unroll
  for (int r = 0; r < 8; ++r) {
    float* o = Hout + (size_t)(rbase + r) * H_DIM + idx15;
    o[0]  = acc0[r];
    o[16] = acc1[r];
    o[32] = acc2[r];
    o[48] = acc3[r];
  }
}

// ---------------------------------------------------------------------------
// Per-node attention scores: e_src[n] = h[n]·a_src, e_dst[n] = h[n]·a_dst.
// One wave32 per node, shuffle reduction.
// ---------------------------------------------------------------------------
__global__ __launch_bounds__(256) void gat_node_scores(
    const float* __restrict__ Hf, const float* __restrict__ a_src,
    const float* __restrict__ a_dst, float* __restrict__ e_src,
    float* __restrict__ e_dst, int N) {
  const int lane = threadIdx.x & 31;
  const int n = (blockIdx.x * blockDim.x + threadIdx.x) >> 5;
  if (n >= N) return;
  float h0 = Hf[(size_t)n * H_DIM + lane];
  float h1 = Hf[(size_t)n * H_DIM + lane + 32];
  float s = h0 * a_src[lane] + h1 * a_src[lane + 32];
  float d = h0 * a_dst[lane] + h1 * a_dst[lane + 32];
#pragma unroll
  for (int off = 16; off > 0; off >>= 1) {
    s += __shfl_xor(s, off, 32);
    d += __shfl_xor(d, off, 32);
  }
  if (lane == 0) { e_src[n] = s; e_dst[n] = d; }
}

// ---------------------------------------------------------------------------
// Per-layer init: out = broadcast(bias); m = -inf; denom = 0.
// ---------------------------------------------------------------------------
__global__ __launch_bounds__(256) void gat_init_layer(
    float* __restrict__ m, float* __restrict__ denom,
    float* __restrict__ out, const float* __restrict__ bias, int N) {
  int i = blockIdx.x * blockDim.x + threadIdx.x;
  if (i < N * H_DIM) out[i] = bias[i & (H_DIM - 1)];
  if (i < N) { m[i] = -INFINITY; denom[i] = 0.0f; }
}

// sign-aware float atomic max (monotone int / reverse-monotone uint encoding)
__device__ __forceinline__ void atomic_max_f32(float* addr, float v) {
  if (v >= 0.0f) atomicMax((int*)addr, __float_as_int(v));
  else           atomicMin((unsigned int*)addr, __float_as_uint(v));
}

// Pass A: e = leaky_relu(e_src[src] + e_dst[dst]); segment max into m[dst].
__global__ __launch_bounds__(256) void gat_edge_a(
    const int* __restrict__ ei, int E, int N,
    const float* __restrict__ e_src, const float* __restrict__ e_dst,
    float* __restrict__ e_edge, float* __restrict__ m) {
  int k = blockIdx.x * blockDim.x + threadIdx.x;
  if (k >= E + N) return;
  int s, d;
  if (k < E) { s = ei[k]; d = ei[E + k]; } else { s = d = k - E; }
  float e = e_src[s] + e_dst[d];
  e = (e >= 0.0f) ? e : NEG_SLOPE * e;
  e_edge[k] = e;
  atomic_max_f32(m + d, e);
}

// Pass B: ex = exp(e - m[dst]); segment sum into denom[dst].
__global__ __launch_bounds__(256) void gat_edge_b(
    const int* __restrict__ ei, int E, int N,
    float* __restrict__ e_edge, const float* __restrict__ m,
    float* __restrict__ denom) {
  int k = blockIdx.x * blockDim.x + threadIdx.x;
  if (k >= E + N) return;
  int d = (k < E) ? ei[E + k] : (k - E);
  float ex = __expf(e_edge[k] - m[d]);
  e_edge[k] = ex;
  atomicAdd(denom + d, ex);
}

// Pass C: out[dst] += (ex / denom[dst]) * h[src].  One wave32 per edge,
// lane handles features {lane, lane+32}.
__global__ __launch_bounds__(256) void gat_edge_c(
    const int* __restrict__ ei, int E, int N,
    const float* __restrict__ e_edge, const float* __restrict__ denom,
    const float* __restrict__ Hf, float* __restrict__ out) {
  const int lane = threadIdx.x & 31;
  const int k = (blockIdx.x * blockDim.x + threadIdx.x) >> 5;
  if (k >= E + N) return;
  int s, d;
  if (k < E) { s = ei[k]; d = ei[E + k]; } else { s = d = k - E; }
  float alpha = e_edge[k] / denom[d];
  float v0 = alpha * Hf[(size_t)s * H_DIM + lane];
  float v1 = alpha * Hf[(size_t)s * H_DIM + lane + 32];
  atomicAdd(out + (size_t)d * H_DIM + lane,      v0);
  atomicAdd(out + (size_t)d * H_DIM + lane + 32, v1);
}

__global__ __launch_bounds__(256) void gat_relu(float* __restrict__ p, int n) {
  int i = blockIdx.x * blockDim.x + threadIdx.x;
  if (i < n) p[i] = fmaxf(p[i], 0.0f);
}

__global__ __launch_bounds__(256) void gat_zero(float* __restrict__ p, int n) {
  int i = blockIdx.x * blockDim.x + threadIdx.x;
  if (i < n) p[i] = 0.0f;
}

// ---------------------------------------------------------------------------
// Mean pool: LDS-blocked accumulation of [B,H] sums + counts. 128 blocks so
// global atomics are only 128 * 8K instead of 6.4M.
// ---------------------------------------------------------------------------
__global__ __launch_bounds__(256) void gat_pool(
    const float* __restrict__ Hf, const int* __restrict__ batch,
    float* __restrict__ psum, float* __restrict__ pcnt, int N) {
  __shared__ float acc[B_GRAPHS * H_DIM];   // 32 KB
  __shared__ float cnt[B_GRAPHS];
  for (int i = threadIdx.x; i < B_GRAPHS * H_DIM; i += blockDim.x) acc[i] = 0.0f;
  for (int i = threadIdx.x; i < B_GRAPHS; i += blockDim.x) cnt[i] = 0.0f;
  __syncthreads();

  int chunk = (N + gridDim.x - 1) / gridDim.x;
  int start = blockIdx.x * chunk;
  int end = min(start + chunk, N);
  int f  = threadIdx.x & (H_DIM - 1);
  int nl = threadIdx.x >> 6;                // 4 nodes in flight per iteration
  for (int n = start + nl; n < end; n += 4) {
    int g = batch[n];
    atomicAdd(&acc[g * H_DIM + f], Hf[(size_t)n * H_DIM + f]);
    if (f == 0) atomicAdd(&cnt[g], 1.0f);
  }
  __syncthreads();
  for (int i = threadIdx.x; i < B_GRAPHS * H_DIM; i += blockDim.x)
    atomicAdd(&psum[i], acc[i]);
  for (int i = threadIdx.x; i < B_GRAPHS; i += blockDim.x)
    atomicAdd(&pcnt[i], cnt[i]);
}

// out[b,c] = (psum[b]/max(cnt,1)) · W_lin[c] + b_lin[c]    (128x10, trivial)
__global__ __launch_bounds__(256) void gat_final_linear(
    const float* __restrict__ psum, const float* __restrict__ pcnt,
    const float* __restrict__ Wl, const float* __restrict__ bl,
    float* __restrict__ out) {
  int i = blockIdx.x * blockDim.x + threadIdx.x;
  if (i >= B_GRAPHS * C_CLASSES) return;
  int b = i / C_CLASSES, c = i % C_CLASSES;
  float inv = 1.0f / fmaxf(pcnt[b], 1.0f);
  float s = bl[c];
#pragma unroll
  for (int k = 0; k < H_DIM; ++k)
    s += psum[b * H_DIM + k] * inv * Wl[c * H_DIM + k];
  out[i] = s;
}

// ---------------------------------------------------------------------------
extern "C" void kernel_launch(void* const* d_in, const int* in_sizes, int n_in,
                              void* d_out, int out_size, void* d_ws, size_t ws_size,
                              hipStream_t stream) {
  const float* x      = (const float*)d_in[0];
  const int*   ei     = (const int*)d_in[1];   // [2,E] (JAX default: int32)
  const int*   batch  = (const int*)d_in[2];
  const float* W1     = (const float*)d_in[3];
  const float* as1    = (const float*)d_in[4];
  const float* ad1    = (const float*)d_in[5];
  const float* b1     = (const float*)d_in[6];
  const float* W2     = (const float*)d_in[7];
  const float* as2    = (const float*)d_in[8];
  const float* ad2    = (const float*)d_in[9];
  const float* b2     = (const float*)d_in[10];
  const float* W3     = (const float*)d_in[11];
  const float* as3    = (const float*)d_in[12];
  const float* ad3    = (const float*)d_in[13];
  const float* b3     = (const float*)d_in[14];
  const float* W_lin  = (const float*)d_in[15];
  const float* b_lin  = (const float*)d_in[16];

  const int E    = in_sizes[1] / 2;
  const int N    = in_sizes[2];
  const int F_IN = in_sizes[0] / N;
  const int EE   = E + N;

  // workspace layout (floats)
  float* bufA   = (float*)d_ws;                 // [N,64] transformed features h
  float* bufB   = bufA + (size_t)N * H_DIM;     // [N,64] layer output / next input
  float* e_edge = bufB + (size_t)N * H_DIM;     // [E+N]
  float* e_src  = e_edge + EE;                  // [N]
  float* e_dst  = e_src + N;                    // [N]
  float* m      = e_dst + N;                    // [N]
  float* denom  = m + N;                        // [N]
  float* psum   = denom + N;                    // [B,64]
  float* pcnt   = psum + B_GRAPHS * H_DIM;      // [B]

  const int TB = 256;
  const int gGemm  = ((N / 16) * 32 + TB - 1) / TB;
  const int gNode  = (N * 32 + TB - 1) / TB;
  const int gInit  = (N * H_DIM + TB - 1) / TB;
  const int gEdge  = (EE + TB - 1) / TB;
  const int gEdgeW = (int)(((size_t)EE * 32 + TB - 1) / TB);
  const int gElem  = (N * H_DIM + TB - 1) / TB;

  auto layer = [&](const float* xin, int K, const float* Wl, const float* as,
                   const float* ad, const float* bl, bool relu) {
    if (K == 128)
      gat_gemm_wmma_f32<128><<<gGemm, TB, 0, stream>>>(xin, Wl, bufA, N);
    else
      gat_gemm_wmma_f32<64><<<gGemm, TB, 0, stream>>>(xin, Wl, bufA, N);
    gat_node_scores<<<gNode, TB, 0, stream>>>(bufA, as, ad, e_src, e_dst, N);
    gat_init_layer<<<gInit, TB, 0, stream>>>(m, denom, bufB, bl, N);
    gat_edge_a<<<gEdge, TB, 0, stream>>>(ei, E, N, e_src, e_dst, e_edge, m);
    gat_edge_b<<<gEdge, TB, 0, stream>>>(ei, E, N, e_edge, m, denom);
    gat_edge_c<<<gEdgeW, TB, 0, stream>>>(ei, E, N, e_edge, denom, bufA, bufB);
    if (relu) gat_relu<<<gElem, TB, 0, stream>>>(bufB, N * H_DIM);
  };

  layer(x,    F_IN,  W1, as1, ad1, b1, true);
  layer(bufB, H_DIM, W2, as2, ad2, b2, true);
  layer(bufB, H_DIM, W3, as3, ad3, b3, false);

  gat_zero<<<(B_GRAPHS * H_DIM + B_GRAPHS + TB - 1) / TB, TB, 0, stream>>>(
      psum, B_GRAPHS * H_DIM + B_GRAPHS);
  gat_pool<<<B_GRAPHS, TB, 0, stream>>>(bufB, batch, psum, pcnt, N);
  gat_final_linear<<<(B_GRAPHS * C_CLASSES + TB - 1) / TB, TB, 0, stream>>>(
      psum, pcnt, W_lin, b_lin, (float*)d_out);
}